// FSMRELoss_61297773249071
// MI455X (gfx1250) — compile-verified
//
#include <hip/hip_runtime.h>
#include <hip/hip_bf16.h>

// Problem constants (from reference): B=8, N=32, L=64
#define BB 8
#define NN 32
#define LL 64
#define SLABS (BB * NN * NN)          // 8192 (b,i,j) slabs
#define SLAB_FLOATS (LL * LL)         // 4096 floats = 16 KB
#define SLABS_PER_BLOCK 8
#define NBLOCKS (SLABS / SLABS_PER_BLOCK)   // 1024 partial triples

#if defined(__HIP_DEVICE_COMPILE__) && defined(__gfx1250__) && \
    __has_builtin(__builtin_amdgcn_global_load_async_to_lds_b128)
#define USE_ASYNC_LDS 1
#else
#define USE_ASYNC_LDS 0
#endif

typedef int v4i_ __attribute__((vector_size(16)));
typedef __attribute__((address_space(1))) v4i_* gptr_v4i;
typedef __attribute__((address_space(3))) v4i_* lptr_v4i;

__device__ __forceinline__ void wait_asynccnt0() {
#if defined(__HIP_DEVICE_COMPILE__) && defined(__gfx1250__)
#if __has_builtin(__builtin_amdgcn_s_wait_asynccnt)
    __builtin_amdgcn_s_wait_asynccnt(0);
#else
    asm volatile("s_wait_asynccnt 0x0" ::: "memory");
#endif
#endif
}

__global__ __launch_bounds__(256) void fsmre_partial_kernel(
    const float* __restrict__ probs,
    const int* __restrict__ target,
    float* __restrict__ partial)  // NBLOCKS * 3 floats
{
    __shared__ __align__(16) float slab[SLAB_FLOATS];   // 16 KB staging buffer
    __shared__ float sCe[8];
    __shared__ int   sCor[8];
    __shared__ int   sAll[8];

    const int tid  = threadIdx.x;
    const int lane = tid & 31;
    const int w    = tid >> 5;          // wave id, 0..7
    const int bid  = blockIdx.x;

    // thread-0 running totals over this block's slabs
    float blkCe    = 0.0f;
    int   blkCor   = 0;
    int   blkMulti = 0;

    for (int s = 0; s < SLABS_PER_BLOCK; ++s) {
        const int slabIdx = bid * SLABS_PER_BLOCK + s;
        // decode (b,i,j); skip diagonal pairs (masked out in reference)
        const int rem = slabIdx % (NN * NN);
        const int i   = rem / NN;
        const int j   = rem % NN;
        if (i == j) continue;   // uniform across block

        // ---- stage 16 KB slab into LDS ----
        const char* gbase = (const char*)probs + (size_t)slabIdx * (SLAB_FLOATS * 4);
#if USE_ASYNC_LDS
        {
            const unsigned off = (unsigned)tid * 16u;
#pragma unroll
            for (int c = 0; c < 4; ++c) {
                __builtin_amdgcn_global_load_async_to_lds_b128(
                    (gptr_v4i)(gbase + off + c * 4096),
                    (lptr_v4i)((char*)slab + off + c * 4096),
                    0, 0);
            }
            wait_asynccnt0();
        }
#else
        {
            float4*       dst = (float4*)slab;
            const float4* src = (const float4*)gbase;
#pragma unroll
            for (int c = 0; c < 4; ++c) dst[tid + c * 256] = src[tid + c * 256];
        }
#endif
        __syncthreads();

        // ---- each wave handles 8 rows of the 64x64 slab ----
        float ceAcc  = 0.0f;
        int   corAcc = 0;
        int   allAcc = 1;
#pragma unroll
        for (int r = 0; r < 8; ++r) {
            const int l = w * 8 + r;                       // row index = label index
            const float2 v = ((const float2*)(slab + l * LL))[lane]; // cols 2*lane, 2*lane+1

            // row max over 64 elements: local pair max + wave32 butterfly
            float m = fmaxf(v.x, v.y);
#pragma unroll
            for (int off = 16; off >= 1; off >>= 1)
                m = fmaxf(m, __shfl_xor(m, off, 32));

            // broadcast diagonal element probs[..,l,l] (held by lane l>>1, comp l&1)
            const float sel  = (l & 1) ? v.y : v.x;
            const float diag = __shfl(sel, l >> 1, 32);

            const int t = target[slabIdx * LL + l];
            const float ce = t ? -__logf(diag) : -log1pf(-diag);
            const int corr = ((diag == m) == (t == 1)) ? 1 : 0;

            if (lane == 0) {
                ceAcc  += ce;
                corAcc += corr;
                allAcc &= corr;
            }
        }
        if (lane == 0) { sCe[w] = ceAcc; sCor[w] = corAcc; sAll[w] = allAcc; }
        __syncthreads();

        if (tid == 0) {
            float ce = 0.0f; int cor = 0, all = 1;
#pragma unroll
            for (int k = 0; k < 8; ++k) { ce += sCe[k]; cor += sCor[k]; all &= sAll[k]; }
            blkCe  += ce;
            blkCor += cor;
            blkMulti += all;
        }
        __syncthreads();   // protect slab/sCe before next iteration restages
    }

    if (tid == 0) {
        partial[bid * 3 + 0] = blkCe;
        partial[bid * 3 + 1] = (float)blkCor;
        partial[bid * 3 + 2] = (float)blkMulti;
    }
}

__global__ __launch_bounds__(256) void fsmre_reduce_kernel(
    const float* __restrict__ partial, float* __restrict__ out)
{
    __shared__ double s0[256], s1[256], s2[256];
    const int tid = threadIdx.x;
    double a = 0.0, b = 0.0, c = 0.0;
    for (int idx = tid; idx < NBLOCKS; idx += 256) {
        a += (double)partial[idx * 3 + 0];
        b += (double)partial[idx * 3 + 1];
        c += (double)partial[idx * 3 + 2];
    }
    s0[tid] = a; s1[tid] = b; s2[tid] = c;
    __syncthreads();
    for (int stride = 128; stride >= 1; stride >>= 1) {
        if (tid < stride) {
            s0[tid] += s0[tid + stride];
            s1[tid] += s1[tid + stride];
            s2[tid] += s2[tid + stride];
        }
        __syncthreads();
    }
    if (tid == 0) {
        const double lossDen   = (double)NN * (NN - 1) * LL * BB;  // 507904
        const double singleDen = lossDen;                          // 507904
        const double multiDen  = (double)BB * NN * (NN - 1);       // 7936
        out[0] = (float)(s0[0] / lossDen);
        out[1] = (float)(s1[0] / singleDen);
        out[2] = (float)(s2[0] / multiDen);
    }
}

extern "C" void kernel_launch(void* const* d_in, const int* in_sizes, int n_in,
                              void* d_out, int out_size, void* d_ws, size_t ws_size,
                              hipStream_t stream) {
    const float* probs  = (const float*)d_in[0];
    const int*   target = (const int*)d_in[1];
    float*       out    = (float*)d_out;
    float*       part   = (float*)d_ws;   // NBLOCKS*3 floats = 12 KB scratch

    fsmre_partial_kernel<<<NBLOCKS, 256, 0, stream>>>(probs, target, part);
    fsmre_reduce_kernel<<<1, 256, 0, stream>>>(part, out);
    (void)in_sizes; (void)n_in; (void)out_size; (void)ws_size;
}